// BatchEmbeddingUpdater_63410897158926
// MI455X (gfx1250) — compile-verified
//
#include <hip/hip_runtime.h>

typedef float v2f __attribute__((ext_vector_type(2)));
typedef float v8f __attribute__((ext_vector_type(8)));
typedef float f4  __attribute__((ext_vector_type(4)));

#define DIM 128
#define LDS_STRIDE 132   // 128 + 4 pad -> conflict-free ds_load_b64 of A-fragments
#define WAVES_PER_BLOCK 4

// ---------------------------------------------------------------------------
// Kernel 1: stream previous_embedding -> out (512 MB, non-temporal)
// ---------------------------------------------------------------------------
__global__ void base_copy_kernel(const f4* __restrict__ src,
                                 f4* __restrict__ dst, long long n4) {
  long long i = (long long)blockIdx.x * blockDim.x + threadIdx.x;
  long long stride = (long long)gridDim.x * blockDim.x;
  for (; i < n4; i += stride) {
    f4 v = __builtin_nontemporal_load(&src[i]);
    __builtin_nontemporal_store(v, &dst[i]);
  }
}

// ---------------------------------------------------------------------------
// Kernel 2: per 16-row tile:
//   H = prev[ids] + X @ Wnig^T + bnig        (GEMM1, WMMA f32 16x16x4)
//   y = H @ Wnode^T + bnode + prev[ids]      (GEMM2, WMMA f32 16x16x4)
//   out[ids] = y                              (scatter)
// One wave per tile. EXEC is always all-ones (WMMA requirement); tail tiles
// are clamped and only the final store is predicated (wave-uniform).
// ---------------------------------------------------------------------------
__global__ __launch_bounds__(WAVES_PER_BLOCK * 32)
void update_scatter_kernel(const int* __restrict__ ids,
                           const float* __restrict__ X,
                           const float* __restrict__ prev,
                           const float* __restrict__ Wnig,
                           const float* __restrict__ bnig,
                           const float* __restrict__ Wnode,
                           const float* __restrict__ bnode,
                           float* __restrict__ out, int nTiles) {
  __shared__ float Hlds[WAVES_PER_BLOCK * 16 * LDS_STRIDE];

  const int lane = threadIdx.x & 31;
  const int wave = threadIdx.x >> 5;
  const int half = lane >> 4;   // which 16-lane half of the wave
  const int lq   = lane & 15;

  int tileId = blockIdx.x * WAVES_PER_BLOCK + wave;
  const bool valid = (tileId < nTiles);
  if (!valid) tileId = nTiles - 1;      // clamp: keep EXEC full, loads in-bounds
  const int b0 = tileId * 16;

  // Node ids for the 8 output rows this lane-half owns (C/D layout: M = v + 8*half)
  int idr[8];
#pragma unroll
  for (int v = 0; v < 8; ++v) idr[v] = ids[b0 + v + 8 * half];

  float* hbase = &Hlds[wave * 16 * LDS_STRIDE];

  // ---------------- GEMM1: H = prev + X @ Wnig^T + bnig ----------------
  // A-fragment (16x4 f32): lane lq holds row m=lq, K = kk + 2*half + {0,1}
  const float* xrow = X + (long long)(b0 + lq) * DIM + 2 * half;
  for (int nt = 0; nt < 8; ++nt) {
    v8f acc = {};
    const float* wrow = Wnig + (long long)(nt * 16 + lq) * DIM + 2 * half;
#pragma unroll
    for (int kk = 0; kk < DIM; kk += 4) {
      v2f a = *reinterpret_cast<const v2f*>(xrow + kk);
      v2f b = *reinterpret_cast<const v2f*>(wrow + kk);
      acc = __builtin_amdgcn_wmma_f32_16x16x4_f32(
          false, a, false, b, (short)0, acc, false, false);
    }
    const int ng = nt * 16 + lq;        // global output column
    const float bb = bnig[ng];
#pragma unroll
    for (int v = 0; v < 8; ++v) {
      // D layout: (M = v + 8*half, N = ng)
      float h = acc[v] + bb + prev[(long long)idr[v] * DIM + ng];
      hbase[(v + 8 * half) * LDS_STRIDE + ng] = h;
    }
  }

  __syncthreads();   // make H visible for A-layout reads

  // ---------------- GEMM2: y = H @ Wnode^T + bnode + prev ----------------
  const float* hrow = hbase + lq * LDS_STRIDE + 2 * half;
  for (int nt = 0; nt < 8; ++nt) {
    v8f acc = {};
    const float* wrow = Wnode + (long long)(nt * 16 + lq) * DIM + 2 * half;
#pragma unroll
    for (int kk = 0; kk < DIM; kk += 4) {
      v2f a = *reinterpret_cast<const v2f*>(hrow + kk);
      v2f b = *reinterpret_cast<const v2f*>(wrow + kk);
      acc = __builtin_amdgcn_wmma_f32_16x16x4_f32(
          false, a, false, b, (short)0, acc, false, false);
    }
    const int ng = nt * 16 + lq;
    const float bb = bnode[ng];
#pragma unroll
    for (int v = 0; v < 8; ++v) {
      long long row = idr[v];
      float y = acc[v] + bb + prev[row * DIM + ng];   // residual +prev
      if (valid) out[row * DIM + ng] = y;             // wave-uniform predicate
    }
  }
}

// ---------------------------------------------------------------------------
extern "C" void kernel_launch(void* const* d_in, const int* in_sizes, int n_in,
                              void* d_out, int out_size, void* d_ws, size_t ws_size,
                              hipStream_t stream) {
  const int*   src_ids = (const int*)d_in[0];
  const int*   dst_ids = (const int*)d_in[1];
  const float* prev    = (const float*)d_in[2];
  const float* Xsrc    = (const float*)d_in[3];
  const float* Xdst    = (const float*)d_in[4];
  const float* Wnig    = (const float*)d_in[5];
  const float* bnig    = (const float*)d_in[6];
  const float* Wnode   = (const float*)d_in[7];
  const float* bnode   = (const float*)d_in[8];
  float*       out     = (float*)d_out;

  // 1) base copy: previous_embedding -> out
  long long n4 = (long long)out_size / 4;
  base_copy_kernel<<<32768, 256, 0, stream>>>((const f4*)prev, (f4*)out, n4);

  // 2) src update+scatter, then 3) dst update+scatter (dst wins conflicts via
  //    stream ordering, matching reference scatter order)
  int batch  = in_sizes[0];
  int nTiles = (batch + 15) / 16;
  int blocks = (nTiles + WAVES_PER_BLOCK - 1) / WAVES_PER_BLOCK;
  update_scatter_kernel<<<blocks, WAVES_PER_BLOCK * 32, 0, stream>>>(
      src_ids, Xsrc, prev, Wnig, bnig, Wnode, bnode, out, nTiles);
  update_scatter_kernel<<<blocks, WAVES_PER_BLOCK * 32, 0, stream>>>(
      dst_ids, Xdst, prev, Wnig, bnig, Wnode, bnode, out, nTiles);
}